// ProposalLayer_33990371181276
// MI455X (gfx1250) — compile-verified
//
#include <hip/hip_runtime.h>

// ---- problem constants (PointRCNN RPN, KITTI Car) ----
#define BB    4
#define NN    16384
#define REGC  76
#define PRE   2250     // RPN_PRE_NMS_TOP_N / B
#define POST  128      // RPN_POST_NMS_TOP_N / B
#define NMS_TH 0.8f

#ifndef __has_builtin
#define __has_builtin(x) 0
#endif

#if __has_builtin(__builtin_amdgcn_global_load_async_to_lds_b32)
#define USE_ASYNC_LDS 1
#else
#define USE_ASYNC_LDS 0
#endif

typedef __attribute__((address_space(1))) int gint;   // global AS pointer elem
typedef __attribute__((address_space(3))) int lint;   // LDS AS pointer elem

__device__ __forceinline__ unsigned int sortable_bits(float f) {
  unsigned int u = __float_as_uint(f);
  return (u & 0x80000000u) ? ~u : (u | 0x80000000u);
}

// ---------------- Kernel 4 (defined first so its asm leads the disasm):
// LDS-resident greedy NMS + output, staged via async global->LDS ----------------
__global__ __launch_bounds__(1024)
void nms_kernel(const float* __restrict__ pr, const float* __restrict__ sc,
                const float* __restrict__ bev, float* __restrict__ out) {
  __shared__ float sbev[5 * PRE];          // 45 KB: x1,x2,z1,z2,area SoA
  __shared__ unsigned char ssup[PRE];
  __shared__ int skeep[POST];
  __shared__ int scount;
  int b = blockIdx.x;
  const float* g = bev + (size_t)b * 5 * PRE;

#if USE_ASYNC_LDS
  {
    gint* gp = (gint*)g;       // cast away const; builtin wants global int*
    lint* lp = (lint*)sbev;    // LDS int*
    for (int t = threadIdx.x; t < 5 * PRE; t += blockDim.x)
      __builtin_amdgcn_global_load_async_to_lds_b32(gp + t, lp + t, 0, 0);
#if __has_builtin(__builtin_amdgcn_s_wait_asynccnt)
    __builtin_amdgcn_s_wait_asynccnt(0);
#else
    asm volatile("s_wait_asynccnt 0" ::: "memory");
#endif
  }
#else
  for (int t = threadIdx.x; t < 5 * PRE; t += blockDim.x) sbev[t] = g[t];
#endif
  for (int t = threadIdx.x; t < PRE; t += blockDim.x) ssup[t] = 0;
  if (threadIdx.x == 0) scount = 0;
  __syncthreads();

  const float* sx1 = sbev;
  const float* sx2 = sbev + PRE;
  const float* sz1 = sbev + 2 * PRE;
  const float* sz2 = sbev + 3 * PRE;
  const float* sar = sbev + 4 * PRE;

  for (int i = 0; i < PRE; ++i) {
    bool keep_i = (ssup[i] == 0);
    if (keep_i) {
      float x1i = sx1[i], x2i = sx2[i], z1i = sz1[i], z2i = sz2[i], ai = sar[i];
      for (int j = i + 1 + (int)threadIdx.x; j < PRE; j += (int)blockDim.x) {
        float iw = fminf(x2i, sx2[j]) - fmaxf(x1i, sx1[j]);
        float ih = fminf(z2i, sz2[j]) - fmaxf(z1i, sz1[j]);
        iw = fmaxf(iw, 0.0f); ih = fmaxf(ih, 0.0f);
        float inter = iw * ih;
        float iou = inter / (ai + sar[j] - inter);
        if (iou > NMS_TH) ssup[j] = 1;
      }
      if (threadIdx.x == 0 && scount < POST) { skeep[scount] = i; scount = scount + 1; }
    }
    __syncthreads();
    if (scount >= POST) break;   // first POST kept are final; later rows irrelevant
  }

  int cnt = scount;
  for (int k = threadIdx.x; k < POST; k += blockDim.x) {
    float box[7] = {0.f, 0.f, 0.f, 0.f, 0.f, 0.f, 0.f};
    float s = 0.0f;
    if (k < cnt) {
      int idx = skeep[k];
      const float* p = pr + ((size_t)b * PRE + idx) * 7;
#pragma unroll
      for (int c = 0; c < 7; ++c) box[c] = p[c];
      s = sc[(size_t)b * PRE + idx];
    }
    float* ob = out + ((size_t)b * POST + k) * 7;
#pragma unroll
    for (int c = 0; c < 7; ++c) ob[c] = box[c];
    out[(size_t)BB * POST * 7 + (size_t)b * POST + k] = s;   // scores after boxes
  }
}

// ---------------- Kernel 1: decode boxes + build sort keys ----------------
__global__ void decode_kernel(const float* __restrict__ scores,
                              const float* __restrict__ reg,
                              const float* __restrict__ xyz,
                              const float* __restrict__ anchor,
                              float* __restrict__ boxes7,
                              unsigned long long* __restrict__ keys) {
  int m = blockIdx.x * blockDim.x + threadIdx.x;
  if (m >= BB * NN) return;
  const float* r = reg + (size_t)m * REGC;
  float px = xyz[m * 3 + 0], py = xyz[m * 3 + 1], pz = xyz[m * 3 + 2];

  int xb = 0; float xv = r[0];
#pragma unroll
  for (int c = 1; c < 12; ++c) { float v = r[c];      if (v > xv) { xv = v; xb = c; } }
  int zb = 0; float zv = r[12];
#pragma unroll
  for (int c = 1; c < 12; ++c) { float v = r[12 + c]; if (v > zv) { zv = v; zb = c; } }

  float pos_x = (float)xb * 0.5f + 0.25f - 3.0f + r[24 + xb] * 0.5f;
  float pos_z = (float)zb * 0.5f + 0.25f - 3.0f + r[36 + zb] * 0.5f;
  float pos_y = py + r[48];

  int rb = 0; float rv = r[49];
#pragma unroll
  for (int c = 1; c < 12; ++c) { float v = r[49 + c]; if (v > rv) { rv = v; rb = c; } }
  float res = r[61 + rb];

  const float APC      = 0.5235987755982988f;  // 2*pi/12
  const float TWO_PI_F = 6.283185307179586f;
  const float PI_F     = 3.141592653589793f;
  float ry = fmodf((float)rb * APC + res * (APC * 0.5f), TWO_PI_F);
  if (ry < 0.0f) ry += TWO_PI_F;          // python floor-mod semantics
  if (ry > PI_F) ry -= TWO_PI_F;

  float a0 = anchor[0], a1 = anchor[1], a2 = anchor[2];
  float h = r[73] * a0 + a0;
  float w = r[74] * a1 + a1;
  float l = r[75] * a2 + a2;

  float* o = boxes7 + (size_t)m * 7;
  o[0] = pos_x + px;
  o[1] = pos_y + 0.5f * h;   // y += h/2
  o[2] = pos_z + pz;
  o[3] = h; o[4] = w; o[5] = l; o[6] = ry;

  unsigned int i_img = (unsigned int)(m & (NN - 1));
  unsigned int s = sortable_bits(scores[m]);
  keys[m] = (((unsigned long long)(~s)) << 32) | i_img;   // asc u64 = desc score, stable
}

// ---------------- Kernel 2a: bitonic global pass (j >= 1024) ----------------
__global__ void bitonic_global(unsigned long long* __restrict__ keys, int k, int j) {
  int t = blockIdx.x * blockDim.x + threadIdx.x;
  if (t >= BB * NN) return;
  int i = t & (NN - 1);
  int b = t >> 14;
  int ixj = i ^ j;
  if (ixj > i) {
    unsigned long long* base = keys + (size_t)b * NN;
    unsigned long long a = base[i], c = base[ixj];
    bool asc = ((i & k) == 0);
    if ((a > c) == asc) { base[i] = c; base[ixj] = a; }
  }
}

// ---------------- Kernel 2b: bitonic LDS-tile pass (all j <= 512 of a stage) --
__global__ __launch_bounds__(512)
void bitonic_shared(unsigned long long* __restrict__ keys, int k, int jstart) {
  __shared__ unsigned long long tile[1024];
  int base = blockIdx.x * 1024;             // tiles never cross image boundaries
  for (int t = threadIdx.x; t < 1024; t += 512) tile[t] = keys[base + t];
  __syncthreads();
  int i_base = base & (NN - 1);             // index within image of tile start
  for (int j = jstart; j > 0; j >>= 1) {
    int t = threadIdx.x;                    // 512 pairs
    int i = ((t & ~(j - 1)) << 1) | (t & (j - 1));
    int ixj = i | j;
    unsigned long long a = tile[i], c = tile[ixj];
    bool asc = (((i_base + i) & k) == 0);
    if ((a > c) == asc) { tile[i] = c; tile[ixj] = a; }
    __syncthreads();
  }
  for (int t = threadIdx.x; t < 1024; t += 512) keys[base + t] = tile[t];
}

// ---------------- Kernel 3: gather top-PRE, build BEV SoA ----------------
__global__ void gather_kernel(const unsigned long long* __restrict__ keys,
                              const float* __restrict__ boxes7,
                              const float* __restrict__ scores,
                              float* __restrict__ pr, float* __restrict__ sc,
                              float* __restrict__ bev) {
  int t = blockIdx.x * blockDim.x + threadIdx.x;
  if (t >= BB * PRE) return;
  int b = t / PRE, rank = t - b * PRE;
  unsigned long long kk = keys[(size_t)b * NN + rank];
  int idx = (int)(unsigned int)(kk & 0xFFFFFFFFull);
  const float* src = boxes7 + ((size_t)b * NN + idx) * 7;
  float bx = src[0], by = src[1], bz = src[2];
  float h = src[3], w = src[4], l = src[5], ry = src[6];
  float* dst = pr + (size_t)t * 7;
  dst[0] = bx; dst[1] = by; dst[2] = bz; dst[3] = h; dst[4] = w; dst[5] = l; dst[6] = ry;
  sc[t] = scores[(size_t)b * NN + idx];
  float x1 = bx - l * 0.5f, x2 = bx + l * 0.5f;
  float z1 = bz - w * 0.5f, z2 = bz + w * 0.5f;
  float* bb = bev + (size_t)b * 5 * PRE;
  bb[0 * PRE + rank] = x1;
  bb[1 * PRE + rank] = x2;
  bb[2 * PRE + rank] = z1;
  bb[3 * PRE + rank] = z2;
  bb[4 * PRE + rank] = (x2 - x1) * (z2 - z1);
}

// ---------------- host-side launch ----------------
extern "C" void kernel_launch(void* const* d_in, const int* in_sizes, int n_in,
                              void* d_out, int out_size, void* d_ws, size_t ws_size,
                              hipStream_t stream) {
  (void)in_sizes; (void)n_in; (void)out_size; (void)ws_size;
  const float* scores = (const float*)d_in[0];
  const float* reg    = (const float*)d_in[1];
  const float* xyz    = (const float*)d_in[2];
  const float* anchor = (const float*)d_in[3];
  float* out = (float*)d_out;

  char* ws = (char*)d_ws;
  size_t off = 0;
  float* boxes7 = (float*)(ws + off);               off += (size_t)BB * NN * 7 * sizeof(float);   // 1,835,008 B
  unsigned long long* keys = (unsigned long long*)(ws + off); off += (size_t)BB * NN * sizeof(unsigned long long);
  float* pr  = (float*)(ws + off);                  off += (size_t)BB * PRE * 7 * sizeof(float);
  float* sc  = (float*)(ws + off);                  off += (size_t)BB * PRE * sizeof(float);
  float* bev = (float*)(ws + off);                  off += (size_t)BB * 5 * PRE * sizeof(float);

  const int total = BB * NN;
  decode_kernel<<<(total + 255) / 256, 256, 0, stream>>>(scores, reg, xyz, anchor, boxes7, keys);

  // bitonic sort: global passes for j>=1024, then one LDS-tile kernel per stage for j<=512
  for (int k = 2; k <= NN; k <<= 1) {
    for (int j = k >> 1; j >= 1024; j >>= 1)
      bitonic_global<<<(total + 255) / 256, 256, 0, stream>>>(keys, k, j);
    int jstart = (k >> 1) < 512 ? (k >> 1) : 512;
    bitonic_shared<<<total / 1024, 512, 0, stream>>>(keys, k, jstart);
  }

  gather_kernel<<<(BB * PRE + 255) / 256, 256, 0, stream>>>(keys, boxes7, scores, pr, sc, bev);
  nms_kernel<<<BB, 1024, 0, stream>>>(pr, sc, bev, out);
}